// AutoDR_33492154974655
// MI455X (gfx1250) — compile-verified
//
#include <hip/hip_runtime.h>

// Problem constants (match reference)
#define N_NODES 50000
#define N_EDGES 800000
#define DIM 64
#define NODE_STRIDE 256        // 4 hops * 64 dims, node-major output layout
#define CHUNK 1000             // edges staged per block (800 blocks exactly)
#define EDGES_PER_WAVE (CHUNK / 8)

typedef unsigned int v4u __attribute__((ext_vector_type(4)));
typedef unsigned int v8u __attribute__((ext_vector_type(8)));

// ---------------------------------------------------------------------------
// CDNA5 async / TDM helpers (gfx1250)
// ---------------------------------------------------------------------------
__device__ __forceinline__ unsigned int lds_offset_of(const void* p) {
    // For LDS-aperture flat addresses, addr[31:0] is the wave-relative LDS
    // byte offset (ISA 10.2 aperture mapping).
    return (unsigned int)(size_t)p;
}

__device__ __forceinline__ void async_load_to_lds_b128(unsigned int lds_off, const void* gaddr) {
    asm volatile("global_load_async_to_lds_b128 %0, %1, off"
                 :: "v"(lds_off), "v"(gaddr) : "memory");
}

__device__ __forceinline__ void wait_asynccnt0() {
    asm volatile("s_wait_asynccnt 0x0" ::: "memory");
}

// 1-D TDM tile load: `count` elements of (1<<data_size_log2) bytes from
// `gaddr` into LDS at byte offset `lds_off`. Descriptor per ISA 08 §8:
//   group0: count=1 | lds_addr | global_addr[56:0] | type=2
//   group1: data_size | tensor_dim0=count | tile_dim0=count | tile_dim1=0 (1-D)
//   groups 2/3: zero (tensor <= 2D)
// All inputs must be workgroup-uniform ("s" constraints -> SGPR groups).
__device__ __forceinline__ void tdm_load_1d(unsigned int lds_off, const void* gaddr,
                                            unsigned int count, unsigned int data_size) {
    const unsigned long long ga = (unsigned long long)(size_t)gaddr;
    v4u g0;
    g0.x = 1u;                                                   // count=1 valid descriptor
    g0.y = lds_off;                                              // lds_addr
    g0.z = (unsigned int)ga;                                     // global_addr[31:0]
    g0.w = (unsigned int)((ga >> 32) & 0x01FFFFFFu) | (2u << 30); // addr[56:32] | type=2
    v8u g1;
    g1.s0 = (data_size << 16);                                   // mask=0, data_size, no flags
    g1.s1 = (count & 0xFFFFu) << 16;                             // tensor_dim0[15:0]
    g1.s2 = ((count >> 16) & 0xFFFFu) | (1u << 16);              // tensor_dim0[31:16], tensor_dim1=1
    g1.s3 = (count & 0xFFFFu) << 16;                             // tile_dim0 = count
    g1.s4 = 0u;                                                  // tile_dim1=0 (1-D), tile_dim2=0
    g1.s5 = count;                                               // tensor_dim0_stride[31:0]
    g1.s6 = (count & 0xFFFFu) << 16;                             // tensor_dim1_stride[15:0] (unused)
    g1.s7 = 0u;
    const v4u gz = {0u, 0u, 0u, 0u};
    asm volatile("tensor_load_to_lds %0, %1, %2, %3"
                 :: "s"(g0), "s"(g1), "s"(gz), "s"(gz) : "memory");
}

// ---------------------------------------------------------------------------
// Kernel 1: out[:,0,:] = embed ; out[:,1:4,:] = 0   (float4 vectorized)
// ---------------------------------------------------------------------------
__global__ __launch_bounds__(256)
void init_out_kernel(const float* __restrict__ embed, float* __restrict__ out) {
    size_t idx = (size_t)blockIdx.x * blockDim.x + threadIdx.x;
    size_t off = idx * 4;                       // float index into out
    size_t n   = off >> 8;                      // node id
    int    r   = (int)(off & 255);              // position within node record
    float4 v = make_float4(0.f, 0.f, 0.f, 0.f);
    if (r < DIM) {
        v = *(const float4*)(embed + n * DIM + r);
    }
    *(float4*)(out + off) = v;
}

// ---------------------------------------------------------------------------
// Kernel 2: one hop of gather-scale-scatter.
// Block = 256 threads (8 waves) owns CHUNK=1000 consecutive edges.
//   - row/col int64 tiles (8 KB each) staged via TDM (wave 0, scalar branch:
//     TDM ignores EXEC, so an EXEC-masked path would issue 8 duplicates)
//   - trend fp32 tile (4 KB) staged via global_load_async_to_lds_b128
// Each wave then walks its 125 edges; each lane covers 2 consecutive dims
// (float2 gather + 2x global_atomic_add_f32 into the L2-resident output).
// ---------------------------------------------------------------------------
__global__ __launch_bounds__(256)
void scatter_hop_kernel(const unsigned long long* __restrict__ rows,
                        const unsigned long long* __restrict__ cols,
                        const float* __restrict__ trend,
                        const float* __restrict__ src,
                        float* __restrict__ dst) {
    __shared__ unsigned long long s_row[CHUNK];
    __shared__ unsigned long long s_col[CHUNK];
    __shared__ __align__(16) float s_t[CHUNK];

    const int tid  = threadIdx.x;
    const int base = blockIdx.x * CHUNK;

    // trend: 250 threads * 16B async copies (overlaps with TDM issue below)
    if (tid < CHUNK / 4) {
        async_load_to_lds_b128(lds_offset_of(&s_t[tid * 4]), trend + base + tid * 4);
    }

    // wave id as a provably uniform scalar -> s_cbranch, single TDM issue
    const int wv = __builtin_amdgcn_readfirstlane(tid) >> 5;
    if (wv == 0) {
        tdm_load_1d(lds_offset_of(&s_row[0]), rows + base, CHUNK, 3u); // 8B elems
        tdm_load_1d(lds_offset_of(&s_col[0]), cols + base, CHUNK, 3u);
        __builtin_amdgcn_s_wait_tensorcnt(0);
    }
    wait_asynccnt0();          // each wave waits for its own trend slice
    __syncthreads();           // publish all LDS tiles block-wide

    const int lane = tid & 31;
    const int d    = lane * 2;              // this lane's 2 dims
    const int e0   = wv * EDGES_PER_WAVE;   // this wave's edge slice

    #pragma unroll 5
    for (int i = 0; i < EDGES_PER_WAVE; ++i) {
        const int e = e0 + i;
        const int r = (int)s_row[e];        // broadcast LDS reads
        const int c = (int)s_col[e];
        const float t = s_t[e];

        const float2 v = *(const float2*)(src + (size_t)r * NODE_STRIDE + d);
        float* dp = dst + (size_t)c * NODE_STRIDE + d;

        unsafeAtomicAdd(dp,     v.x * t);   // global_atomic_add_f32
        unsafeAtomicAdd(dp + 1, v.y * t);
    }
}

// ---------------------------------------------------------------------------
// Launch
// ---------------------------------------------------------------------------
extern "C" void kernel_launch(void* const* d_in, const int* in_sizes, int n_in,
                              void* d_out, int out_size, void* d_ws, size_t ws_size,
                              hipStream_t stream) {
    const float* embed = (const float*)d_in[0];
    const unsigned long long* eidx = (const unsigned long long*)d_in[1]; // int64 (2, N_EDGES)
    const float* trend = (const float*)d_in[2];
    float* out = (float*)d_out;

    const unsigned long long* rows = eidx;            // edge_index[0]
    const unsigned long long* cols = eidx + N_EDGES;  // edge_index[1]

    // hop 0 copy + zero hops 1..3 (atomic targets must start at 0 every call)
    {
        const size_t total_f4 = (size_t)N_NODES * NODE_STRIDE / 4; // 3.2M
        const int blocks = (int)((total_f4 + 255) / 256);          // 12500
        init_out_kernel<<<blocks, 256, 0, stream>>>(embed, out);
    }

    // hops 1..3: stream-ordered launches give the required global sync
    const int edge_blocks = N_EDGES / CHUNK;                       // 800
    for (int hop = 1; hop <= 3; ++hop) {
        scatter_hop_kernel<<<edge_blocks, 256, 0, stream>>>(
            rows, cols, trend,
            out + (size_t)(hop - 1) * DIM,
            out + (size_t)hop * DIM);
    }
}